// AtomEmbedding_53953379172378
// MI455X (gfx1250) — compile-verified
//
#include <hip/hip_runtime.h>
#include <stdint.h>

// Problem geometry (from reference)
#define N_BATCH        9000
#define ATOMS_PER_MOL  64
#define N_ROWS         (N_BATCH * ATOMS_PER_MOL)   // 576000 atoms
#define EMBED          64
#define NTYPES         100
#define TBL_FLOATS     (NTYPES * EMBED)            // 6400 floats = 25.6 KB
#define TBL_CHUNKS     (TBL_FLOATS / 4)            // 1600 float4
#define CHUNKS_PER_ROW (EMBED / 4)                 // 16 float4 per output row
#define TOTAL_CHUNKS   (N_ROWS * CHUNKS_PER_ROW)   // 9,216,000 float4

typedef __attribute__((ext_vector_type(4))) float v4f;  // native vector: OK for nontemporal builtins

// ---------------------------------------------------------------------------
// K1: zero the 100-bit presence bitmap (4 u32 in d_ws). ws is poisoned once
// by the harness and never re-poisoned, so we must re-init on every call.
// ---------------------------------------------------------------------------
__global__ void k_init(uint32_t* __restrict__ bitmap) {
    if (threadIdx.x < 4) bitmap[threadIdx.x] = 0u;
}

// ---------------------------------------------------------------------------
// K2: presence bitmap. LDS-local atomicOr per block, 4 global atomics/block.
// ---------------------------------------------------------------------------
__global__ void __launch_bounds__(256)
k_presence(const int* __restrict__ types, uint32_t* __restrict__ bitmap, int n) {
    __shared__ uint32_t lbm[4];
    if (threadIdx.x < 4) lbm[threadIdx.x] = 0u;
    __syncthreads();
    const int stride = gridDim.x * blockDim.x;
    for (int i = blockIdx.x * blockDim.x + threadIdx.x; i < n; i += stride) {
        const int v = types[i];                 // 0..99
        atomicOr(&lbm[v >> 5], 1u << (v & 31));
    }
    __syncthreads();
    if (threadIdx.x < 4 && lbm[threadIdx.x] != 0u)
        atomicOr(&bitmap[threadIdx.x], lbm[threadIdx.x]);
}

// ---------------------------------------------------------------------------
// K3: build rank-permuted table  T[v][:] = embedding[rank(v)][:]
// rank(v) = popcount of presence bits below v  (== searchsorted(uniq, v)).
// Single block; 6400 elements.
// ---------------------------------------------------------------------------
__global__ void __launch_bounds__(256)
k_build_table(const uint32_t* __restrict__ bitmap,
              const float* __restrict__ embedding,
              float* __restrict__ T) {
    __shared__ uint32_t bm[4];
    if (threadIdx.x < 4) bm[threadIdx.x] = bitmap[threadIdx.x];
    __syncthreads();
    for (int idx = threadIdx.x; idx < TBL_FLOATS; idx += blockDim.x) {
        const int v = idx >> 6;                 // type value 0..99
        const int d = idx & 63;                 // embed dim
        const int w = v >> 5, b = v & 31;
        uint32_t r = 0;
        #pragma unroll
        for (int j = 0; j < 4; ++j)
            if (j < w) r += __popc(bm[j]);
        r += __popc(bm[w] & ((1u << b) - 1u));  // b==0 -> mask 0
        T[idx] = embedding[(int)r * EMBED + d];
    }
}

// ---------------------------------------------------------------------------
// K4: main gather. Stage T into LDS with CDNA5 async global->LDS b128 copies
// (ASYNCcnt path), then stream output: 1 LDS b128 read + 1 global NT b128
// store per chunk. 16 consecutive lanes share one atom_types dword.
// ---------------------------------------------------------------------------
__global__ void __launch_bounds__(256)
k_gather(const int* __restrict__ types,
         const float* __restrict__ T,          // 100x64 f32 in global scratch
         v4f* __restrict__ out) {
    __shared__ v4f ldsT[TBL_CHUNKS];           // 25.6 KB

    // --- async stage: global T -> LDS, 16B per lane per issue ---
    {
        const uint64_t gbase = (uint64_t)(const void*)T;
        for (int c = threadIdx.x; c < TBL_CHUNKS; c += blockDim.x) {
            const uint32_t lds_addr = (uint32_t)(uintptr_t)(&ldsT[c]); // LDS byte offset (flat low 32b)
            const uint64_t gaddr    = gbase + (uint64_t)c * 16u;
            asm volatile("global_load_async_to_lds_b128 %0, %1, off"
                         :: "v"(lds_addr), "v"(gaddr)
                         : "memory");
        }
        asm volatile("s_wait_asynccnt 0x0" ::: "memory");
    }
    __syncthreads();

    // --- streaming gather: NT stores keep the 147.5 MB output stream from
    //     thrashing L2 (table + type stream stay L2-hot) ---
    const int tid    = blockIdx.x * blockDim.x + threadIdx.x;
    const int stride = gridDim.x * blockDim.x;
    for (int c = tid; c < TOTAL_CHUNKS; c += stride) {
        const int row = c >> 4;                 // atom index
        const int sub = c & 15;                 // float4 chunk within row
        // prefetch the type stream ahead (global_prefetch_b8)
        const int pf = row + 4096;
        if (pf < N_ROWS) __builtin_prefetch(types + pf, 0, 0);
        const int t = __builtin_nontemporal_load(types + row);   // 0..99
        const v4f v = ldsT[t * CHUNKS_PER_ROW + sub];
        __builtin_nontemporal_store(v, out + c);
    }
}

// ---------------------------------------------------------------------------
extern "C" void kernel_launch(void* const* d_in, const int* in_sizes, int n_in,
                              void* d_out, int out_size, void* d_ws, size_t ws_size,
                              hipStream_t stream) {
    const int*   atom_types = (const int*)d_in[0];     // [9000*64] int32
    const float* embedding  = (const float*)d_in[1];   // [100*64]  f32
    v4f*         out        = (v4f*)d_out;             // [9000*64*64] f32

    uint32_t* bitmap = (uint32_t*)d_ws;                       // 16 B
    float*    T      = (float*)((char*)d_ws + 256);           // 25.6 KB, 256B aligned

    k_init<<<1, 32, 0, stream>>>(bitmap);
    k_presence<<<512, 256, 0, stream>>>(atom_types, bitmap, N_ROWS);
    k_build_table<<<1, 256, 0, stream>>>(bitmap, embedding, T);
    k_gather<<<1024, 256, 0, stream>>>(atom_types, T, out);
}